// PositionAttentionModule3_5523327942728
// MI455X (gfx1250) — compile-verified
//
#include <hip/hip_runtime.h>
#include <hip/hip_bf16.h>
#include <cstdint>

// Problem constants (from reference): n=8, c=64, h=w=256, points=8192, 5x5 taps.
#define NB   8
#define CC   64
#define HH   256
#define WW   256
#define PTS  8192
#define HWs  65536            // H*W
#define NELEM 33554432LL      // n*c*h*w
#define NPOINT_WAVES 65536    // n*points, one wave32 per point

typedef __attribute__((ext_vector_type(2))) float v2f;
typedef __attribute__((ext_vector_type(8))) float v8f;

// ---------------------------------------------------------------------------
// Kernel A: x_out = (1 + alpha) * x1 everywhere (bandwidth bound, float4).
// ---------------------------------------------------------------------------
__global__ __launch_bounds__(256)
void pam_scale_kernel(const float* __restrict__ x1,
                      const float* __restrict__ alpha,
                      float* __restrict__ out, long long n4) {
  const float s = 1.0f + alpha[0];
  long long i = (long long)blockIdx.x * blockDim.x + threadIdx.x;
  const long long stride = (long long)gridDim.x * blockDim.x;
  const float4* __restrict__ in4 = (const float4*)x1;
  float4* __restrict__ out4 = (float4*)out;
  for (; i < n4; i += stride) {
    float4 v = in4[i];
    v.x *= s; v.y *= s; v.z *= s; v.w *= s;
    out4[i] = v;
  }
}

// ---------------------------------------------------------------------------
// Kernel B: per-point attention. One wave32 per point, 4 waves per block.
// LDS per wave: xa 32x66 (25 real tap rows + 7 zero pad rows, row-padded to
// 66 floats for bank-conflict-free A-fragment reads) + xp 64 + energy 32.
// ---------------------------------------------------------------------------
#define XA_STRIDE 66
#define WAVE_LDS  (32*XA_STRIDE + 64 + 32)   // 2208 floats

__device__ __forceinline__ int reflect_idx(int i) {
  i = (i < 0) ? -i : i;                // reflect low side
  i = (i >= HH) ? (2*HH - 2 - i) : i;  // reflect high side
  return i;
}

__global__ __launch_bounds__(128)
void pam_point_kernel(const float* __restrict__ x1,
                      const float* __restrict__ x2,
                      const int* __restrict__ SFL,
                      const float* __restrict__ alpha_p,
                      float* __restrict__ out) {
  __shared__ float smem[4 * WAVE_LDS];
  const int lane = threadIdx.x & 31;
  const int wv   = threadIdx.x >> 5;
  float* xa = smem + wv * WAVE_LDS;        // [32][66]
  float* xp = xa + 32 * XA_STRIDE;         // [64]
  float* en = xp + 64;                     // [32]

  const int gwave = blockIdx.x * 4 + wv;   // 0..65535, exact fit
  const int b = gwave >> 13;               // / 8192
  const int p = gwave & (PTS - 1);
  const int ph = SFL[(b * PTS + p) * 2 + 0];
  const int pw = SFL[(b * PTS + p) * 2 + 1];
  const long long baseB = (long long)b * CC * HWs;
  const int pix = ph * WW + pw;

  // ---- Phase 1: async gathers into LDS (per-lane scatter/gather path) ----
  // xp: x1[b, c, ph, pw] for c = lane, lane+32
  {
    unsigned l0 = (unsigned)(uintptr_t)(&xp[lane]);
    unsigned l1 = (unsigned)(uintptr_t)(&xp[lane + 32]);
    unsigned long long g0 =
        (unsigned long long)(uintptr_t)(x1 + baseB + (long long)lane * HWs + pix);
    unsigned long long g1 =
        (unsigned long long)(uintptr_t)(x1 + baseB + (long long)(lane + 32) * HWs + pix);
    asm volatile("global_load_async_to_lds_b32 %0, %1, off"
                 :: "v"(l0), "v"(g0) : "memory");
    asm volatile("global_load_async_to_lds_b32 %0, %1, off"
                 :: "v"(l1), "v"(g1) : "memory");
  }
  // xa: x2 reflect-padded 5x5 neighborhood, all 64 channels (1600 values)
  for (int it = 0; it < 50; ++it) {
    const int idx = it * 32 + lane;        // 0..1599
    const int k = idx >> 6;                // tap 0..24 (di=k/5, dj=k%5)
    const int c = idx & 63;
    const int rh = reflect_idx(ph + (k / 5) - 2);
    const int rw = reflect_idx(pw + (k % 5) - 2);
    unsigned l = (unsigned)(uintptr_t)(&xa[k * XA_STRIDE + c]);
    unsigned long long g = (unsigned long long)(uintptr_t)
        (x2 + baseB + (long long)c * HWs + (long long)(rh * WW + rw));
    asm volatile("global_load_async_to_lds_b32 %0, %1, off"
                 :: "v"(l), "v"(g) : "memory");
  }
  // zero pad rows 25..31 (cols 0..63) so M-tile 1 of the WMMA sees zeros
  for (int it = 0; it < 14; ++it) {
    const int idx = it * 32 + lane;        // 0..447
    xa[(25 + (idx >> 6)) * XA_STRIDE + (idx & 63)] = 0.0f;
  }
  asm volatile("s_wait_asynccnt 0x0" ::: "memory");
  __syncthreads();

  // ---- Phase 2: energy[k] = sum_c xa[k][c]*xp[c] via V_WMMA_F32_16X16X4_F32
  // A tile: 16 tap-rows x 4 channels; B: 4 channels broadcast into all 16
  // columns (every D column equals the energy). Two M-tiles cover 25 taps.
  // Fragment layout (32-bit A 16x4): lanes 0-15 hold K=0,1; lanes 16-31 K=2,3.
  const int sel = (lane >= 16) ? 2 : 0;
  const int row = lane & 15;
  v8f acc0 = {};   // taps 0..15
  v8f acc1 = {};   // taps 16..24 (+ zero pad rows)
  for (int i = 0; i < 16; ++i) {
    const int cb = 4 * i;
    v2f a0 = *(const v2f*)(&xa[row * XA_STRIDE + cb + sel]);
    v2f a1 = *(const v2f*)(&xa[(16 + row) * XA_STRIDE + cb + sel]);
    v2f bb = *(const v2f*)(&xp[cb + sel]);
    acc0 = __builtin_amdgcn_wmma_f32_16x16x4_f32(
        false, a0, false, bb, (short)0, acc0, false, false);
    acc1 = __builtin_amdgcn_wmma_f32_16x16x4_f32(
        false, a1, false, bb, (short)0, acc1, false, false);
  }

  // D layout: VGPR r, lanes 0-15 -> M=r ; lanes 16-31 -> M=r+8 (all N equal).
  if (lane == 0) {
    for (int r = 0; r < 8; ++r) { en[r] = acc0[r]; en[16 + r] = acc1[r]; }
  } else if (lane == 16) {
    for (int r = 0; r < 8; ++r) { en[8 + r] = acc0[r]; }
    en[24] = acc1[0];
  }
  __syncthreads();

  // ---- Phase 3: softmax over 25 taps (redundant per lane) + fusion ----
  float m = -3.402823466e38f;
  for (int k = 0; k < 25; ++k) m = fmaxf(m, en[k]);
  float ssum = 0.0f;
  for (int k = 0; k < 25; ++k) ssum += __expf(en[k] - m);
  const float inv = 1.0f / ssum;

  float f0 = 0.0f, f1 = 0.0f;   // fusion for channels c=lane, c=lane+32
  for (int k = 0; k < 25; ++k) {
    const float w = __expf(en[k] - m) * inv;
    f0 += w * xa[k * XA_STRIDE + lane];
    f1 += w * xa[k * XA_STRIDE + 32 + lane];
  }

  // ---- Phase 4: point-pixel writeback: out = x1 + alpha * fusion ----
  const float al = alpha_p[0];
  const long long o0 = baseB + (long long)lane * HWs + pix;
  const long long o1 = baseB + (long long)(lane + 32) * HWs + pix;
  out[o0] = x1[o0] + al * f0;
  out[o1] = x1[o1] + al * f1;
}

// ---------------------------------------------------------------------------
extern "C" void kernel_launch(void* const* d_in, const int* in_sizes, int n_in,
                              void* d_out, int out_size, void* d_ws, size_t ws_size,
                              hipStream_t stream) {
  const float* x1    = (const float*)d_in[0];
  const float* x2    = (const float*)d_in[1];
  const int*   SFL   = (const int*)d_in[2];
  const float* alpha = (const float*)d_in[3];
  float* out = (float*)d_out;

  // Kernel A: elementwise (1+alpha)*x1 over all 33.5M elements (float4).
  pam_scale_kernel<<<8192, 256, 0, stream>>>(x1, alpha, out, NELEM / 4);

  // Kernel B: 65536 points, one wave32 each, 4 waves per 128-thread block.
  pam_point_kernel<<<NPOINT_WAVES / 4, 128, 0, stream>>>(x1, x2, SFL, alpha, out);

  // Second tuple output: SFL passthrough (raw 4-byte copy into tail).
  hipMemcpyAsync((char*)d_out + (size_t)NELEM * sizeof(float), (const void*)SFL,
                 (size_t)(NB * PTS * 2) * sizeof(int),
                 hipMemcpyDeviceToDevice, stream);
}